// PositionMultiHeadAttention_6451040878651
// MI455X (gfx1250) — compile-verified
//
#include <hip/hip_runtime.h>
#include <hip/hip_bf16.h>
#include <math.h>

// Problem constants (from reference setup_inputs)
#define TGT   512
#define BSZ   16
#define EMB   1024
#define HEADS 16
#define HD    64
#define SS    32
#define R     4          // attention rows per workgroup
#define NEGM  -1000000.0f
#define LDSTR 516        // 512 + 4 pad: consecutive rows land 4 banks apart -> conflict-free

typedef float v2f __attribute__((ext_vector_type(2)));
typedef float v8f __attribute__((ext_vector_type(8)));
typedef int   v4i __attribute__((ext_vector_type(4)));

// exact parameter types of __builtin_amdgcn_global_load_async_to_lds_b128
// (from clang diagnostic: int4 in AS1 / AS3, non-const)
#define AS_G(p) ((__attribute__((address_space(1))) v4i*)(p))
#define AS_L(p) ((__attribute__((address_space(3))) v4i*)(p))

__device__ __forceinline__ v8f wmma_f32_4(v2f a, v2f b, v8f c) {
    // D = A(16x4,f32) x B(4x16,f32) + C(16x16,f32); exact fp32 via tensor pipe
    return __builtin_amdgcn_wmma_f32_16x16x4_f32(false, a, false, b, (short)0, c,
                                                 false, false);
}

// ---------------------------------------------------------------------------
// GEMM: Y(8192x1024) = X(8192x1024) @ W^T(1024x1024) + bias
// One wave computes a 16x64 tile: 4 independent WMMA accumulator chains that
// share one A fetch per k-step (hides WMMA D->C latency, 4x A reuse).
// MODE 0: scatter output to v workspace laid out (b*H+h, t, hd)
// MODE 1: row-major output (attn_output)
// ---------------------------------------------------------------------------
template <int MODE>
__global__ __launch_bounds__(256) void gemm_bias_kernel(
    const float* __restrict__ X, const float* __restrict__ W,
    const float* __restrict__ bias, float* __restrict__ Y) {
    const int lane = threadIdx.x & 31;
    const int wave = threadIdx.x >> 5;
    const int m0   = blockIdx.x * 16;
    const int n0   = (blockIdx.y * 8 + wave) * 64;   // 4 x 16 columns per wave
    const int mn   = lane & 15;            // row (A) / col (B,C) index
    const int koff = (lane >> 4) * 2;      // upper half-wave supplies K+2,K+3

    const float* xrow = X + (size_t)(m0 + mn) * EMB;
    const float* wrow[4];
#pragma unroll
    for (int nt = 0; nt < 4; ++nt)         // B[k][n] = W[n][k]
        wrow[nt] = W + (size_t)(n0 + nt * 16 + mn) * EMB;

    v8f c[4];
#pragma unroll
    for (int nt = 0; nt < 4; ++nt) c[nt] = (v8f){0.f,0.f,0.f,0.f,0.f,0.f,0.f,0.f};

    for (int k0 = 0; k0 < EMB; k0 += 4) {
        v2f a = *(const v2f*)(xrow + k0 + koff);
#pragma unroll
        for (int nt = 0; nt < 4; ++nt) {
            v2f b = *(const v2f*)(wrow[nt] + k0 + koff);
            c[nt] = wmma_f32_4(a, b, c[nt]);   // 4 independent chains
        }
    }

    const int rbase = (lane >> 4) * 8;     // C layout: VGPR i = row i (lo) / i+8 (hi)
#pragma unroll
    for (int nt = 0; nt < 4; ++nt) {
        const int   ge = n0 + nt * 16 + mn;
        const float bv = bias[ge];
#pragma unroll
        for (int i = 0; i < 8; ++i) {
            const int   gm  = m0 + rbase + i;  // row in (t*16+b) space
            const float val = c[nt][i] + bv;
            if (MODE == 0) {
                const int t = gm >> 4, bb = gm & 15;
                const int h = ge >> 6, d  = ge & 63;
                Y[((size_t)(bb * HEADS + h) * TGT + t) * HD + d] = val;
            } else {
                Y[(size_t)gm * EMB + ge] = val;
            }
        }
    }
}

// ---------------------------------------------------------------------------
// Fused attention: per (batch b, 4-row tile), 16 waves = 16 heads.
//   phase 1: adj tile streamed HBM->LDS by wave 0 via async-to-LDS engine
//            (double-buffered, ASYNCcnt), all 16 heads accumulate aw from LDS
//   phase 2: +bias, +mask, softmax (wave32 shfl reductions) -> probs in LDS
//   phase 3: avg_weights = mean over heads (deterministic, no atomics)
//   phase 4: P(4x512, padded M=16) @ v_h(512x64), 4 independent WMMA chains
// Dynamic LDS: probs 129KB + adj double buffer 16KB = 145KB
// ---------------------------------------------------------------------------
#define ADJ_TILE (R * TGT)                 // 2048 floats = 8KB per s-slice

__global__ __launch_bounds__(512) void attn_fused_kernel(
    const float* __restrict__ adj, const unsigned char* __restrict__ mask,
    const float* __restrict__ posW, const float* __restrict__ posB,
    const float* __restrict__ v_ws, float* __restrict__ attnI,
    float* __restrict__ avg) {
    extern __shared__ float lds[];
    float* adjbuf = lds + HEADS * R * LDSTR;   // 2 x ADJ_TILE floats
    const int lane = threadIdx.x & 31;
    const int h    = threadIdx.x >> 5;     // wave index == head
    const int b    = blockIdx.y;
    const int row0 = blockIdx.x * R;

    const float pw_l = posW[h * SS + lane];  // lane l holds posW[h][s=l]
    const float pb   = posB[h];

    // lane owns cols {j*128 + lane*4 + q : j<4, q<4} (conflict-free float4 tiles)
    float acc[R][16];
#pragma unroll
    for (int r = 0; r < R; ++r)
#pragma unroll
        for (int i = 0; i < 16; ++i) acc[r][i] = 0.f;

    // ---- phase 1: aw = sum_s posW[h][s] * adj[b][s][row][col] ----
    // wave 0 streams the 8KB (R x 512) slice for each s via async-to-LDS.
    const size_t adj_base = (((size_t)b * SS) * TGT + row0) * TGT;
    if (h == 0) {  // prologue: slice s=0 into buffer 0
        const char* g = (const char*)(adj + adj_base);
        char*       l = (char*)adjbuf;
#pragma unroll
        for (int i = 0; i < 16; ++i)
            __builtin_amdgcn_global_load_async_to_lds_b128(
                AS_G(g + lane * 16 + i * 512),
                AS_L(l + lane * 16 + i * 512), 0, 0);
        __builtin_amdgcn_s_wait_asynccnt(0);
    }
    __syncthreads();

    for (int s = 0; s < SS; ++s) {
        if (h == 0 && s + 1 < SS) {  // prefetch next slice into other buffer
            const char* g = (const char*)(adj + adj_base + (size_t)(s + 1) * TGT * TGT);
            char*       l = (char*)(adjbuf + ((s + 1) & 1) * ADJ_TILE);
#pragma unroll
            for (int i = 0; i < 16; ++i)
                __builtin_amdgcn_global_load_async_to_lds_b128(
                    AS_G(g + lane * 16 + i * 512),
                    AS_L(l + lane * 16 + i * 512), 0, 0);
        }

        const float  ws   = __shfl(pw_l, s, 32);
        const float* tile = adjbuf + (s & 1) * ADJ_TILE;
#pragma unroll
        for (int r = 0; r < R; ++r) {
            const float* rp = tile + r * TGT;
#pragma unroll
            for (int j = 0; j < 4; ++j) {
                const float4 x = *(const float4*)(rp + j * 128 + lane * 4);
                acc[r][j * 4 + 0] += ws * x.x;
                acc[r][j * 4 + 1] += ws * x.y;
                acc[r][j * 4 + 2] += ws * x.z;
                acc[r][j * 4 + 3] += ws * x.w;
            }
        }

        if (h == 0) __builtin_amdgcn_s_wait_asynccnt(0);
        __syncthreads();   // next slice resident; everyone done with this one
    }

    // ---- phase 2: mask + softmax, probs -> LDS ----
#pragma unroll
    for (int r = 0; r < R; ++r) {
        const unsigned char* mrow = mask + ((size_t)b * TGT + row0 + r) * TGT;
        float x[16];
#pragma unroll
        for (int j = 0; j < 4; ++j) {
            const uchar4 mk = *(const uchar4*)(mrow + j * 128 + lane * 4);
            x[j * 4 + 0] = acc[r][j * 4 + 0] + pb + (mk.x ? NEGM : 0.f);
            x[j * 4 + 1] = acc[r][j * 4 + 1] + pb + (mk.y ? NEGM : 0.f);
            x[j * 4 + 2] = acc[r][j * 4 + 2] + pb + (mk.z ? NEGM : 0.f);
            x[j * 4 + 3] = acc[r][j * 4 + 3] + pb + (mk.w ? NEGM : 0.f);
        }
        float mx = x[0];
#pragma unroll
        for (int i = 1; i < 16; ++i) mx = fmaxf(mx, x[i]);
        for (int off = 16; off > 0; off >>= 1)
            mx = fmaxf(mx, __shfl_xor(mx, off, 32));
        float sum = 0.f;
#pragma unroll
        for (int i = 0; i < 16; ++i) { x[i] = __expf(x[i] - mx); sum += x[i]; }
        for (int off = 16; off > 0; off >>= 1) sum += __shfl_xor(sum, off, 32);
        const float inv = (sum > 0.f) ? 1.f / sum : 0.f;  // fix_nans equivalent

        float* lrow = lds + (h * R + r) * LDSTR;
#pragma unroll
        for (int j = 0; j < 4; ++j) {
            const int c0 = j * 128 + lane * 4;
            float4 pv = make_float4(x[j * 4 + 0] * inv, x[j * 4 + 1] * inv,
                                    x[j * 4 + 2] * inv, x[j * 4 + 3] * inv);
            *(float4*)(lrow + c0) = pv;
        }
    }
    __syncthreads();

    // ---- phase 3: avg_weights = (1/H) * sum_h P ----
    {
        const int c = threadIdx.x;  // 512 threads == 512 columns
#pragma unroll
        for (int r = 0; r < R; ++r) {
            float ssum = 0.f;
            for (int hh = 0; hh < HEADS; ++hh)
                ssum += lds[(hh * R + r) * LDSTR + c];
            avg[((size_t)b * TGT + row0 + r) * TGT + c] = ssum * (1.0f / HEADS);
        }
    }

    // ---- phase 4: out_h(4x64) = P_h(4x512) @ v_h(512x64) ----
    // Single k-loop, one LDS A fetch feeds 4 independent WMMA chains (nt tiles).
    const float* vbase = v_ws + (size_t)(b * HEADS + h) * TGT * HD;
    const int    m     = lane & 15;
    const int    m4    = m & (R - 1);        // duplicate rows >=R; D rows discarded
    const int    koff  = (lane >> 4) * 2;
    const float* arow  = lds + (h * R + m4) * LDSTR;

    v8f cc[4];
#pragma unroll
    for (int nt = 0; nt < 4; ++nt) cc[nt] = (v8f){0.f,0.f,0.f,0.f,0.f,0.f,0.f,0.f};

    for (int k0 = 0; k0 < TGT; k0 += 4) {
        v2f a = *(const v2f*)(arow + k0 + koff);
        const float* vk0 = vbase + (size_t)(k0 + koff) * HD + m;
#pragma unroll
        for (int nt = 0; nt < 4; ++nt) {
            v2f bv;
            bv.x = vk0[nt * 16];
            bv.y = vk0[nt * 16 + HD];
            cc[nt] = wmma_f32_4(a, bv, cc[nt]);
        }
    }
    if (lane < 16) {  // rows 0..R-1 live in cc[nt][0..R-1] of lanes 0..15
#pragma unroll
        for (int nt = 0; nt < 4; ++nt)
#pragma unroll
            for (int i = 0; i < R; ++i)
                attnI[(size_t)(row0 + i) * (BSZ * EMB) + b * EMB + h * HD +
                      nt * 16 + m] = cc[nt][i];
    }
}

// ---------------------------------------------------------------------------
extern "C" void kernel_launch(void* const* d_in, const int* in_sizes, int n_in,
                              void* d_out, int out_size, void* d_ws, size_t ws_size,
                              hipStream_t stream) {
    const float*         value = (const float*)d_in[0];
    const float*         adj   = (const float*)d_in[1];
    const unsigned char* mask  = (const unsigned char*)d_in[2];  // jnp.bool_ = 1 byte
    const float*         inW   = (const float*)d_in[3];
    const float*         inB   = (const float*)d_in[4];
    const float*         outW  = (const float*)d_in[5];
    const float*         outB  = (const float*)d_in[6];
    const float*         posW  = (const float*)d_in[7];
    const float*         posB  = (const float*)d_in[8];

    float* out   = (float*)d_out;                      // (tgt, bsz, E)
    float* avg   = out + (size_t)TGT * BSZ * EMB;      // (bsz, tgt, tgt)
    float* v_ws  = (float*)d_ws;                       // (b*H+h, t, hd): 32 MB
    float* attnI = v_ws + (size_t)BSZ * HEADS * TGT * HD;  // (t,b,E): 32 MB

    // K1: v = value @ inW^T + inB  -> v workspace
    dim3 g1(TGT * BSZ / 16, EMB / (8 * 64));
    hipLaunchKernelGGL((gemm_bias_kernel<0>), g1, dim3(256), 0, stream,
                       value, inW, inB, v_ws);

    // K2: fused positional-bias attention
    const int smem = (HEADS * R * LDSTR + 2 * ADJ_TILE) * (int)sizeof(float);
    (void)hipFuncSetAttribute((const void*)attn_fused_kernel,
                              hipFuncAttributeMaxDynamicSharedMemorySize, smem);
    dim3 g2(TGT / R, BSZ);
    hipLaunchKernelGGL(attn_fused_kernel, g2, dim3(512), smem, stream,
                       adj, mask, posW, posB, v_ws, attnI, avg);

    // K3: attn_output = attnI @ outW^T + outB
    hipLaunchKernelGGL((gemm_bias_kernel<1>), g1, dim3(256), 0, stream,
                       attnI, outW, outB, out);
}